// GNN_30288109371597
// MI455X (gfx1250) — compile-verified
//
#include <hip/hip_runtime.h>
#include <hip/hip_bf16.h>

typedef float v2f __attribute__((ext_vector_type(2)));
typedef float v8f __attribute__((ext_vector_type(8)));

#define D 128
#define NNODES 100000
#define NEDGES 1600000
#define NGRAPH 64
#define NOUT 64
#define LRELU(x) ((x) > 0.0f ? (x) : 0.01f * (x))

// ---------------------------------------------------------------- utilities
__global__ void zero_f32(float* __restrict__ p, long long n) {
    long long t = (long long)blockIdx.x * blockDim.x + threadIdx.x;
    if (t < n) p[t] = 0.0f;
}

// out = a + s*b  (s read from device scalar pointer)
__global__ void add_scaled(const float* __restrict__ a, const float* __restrict__ b,
                           const float* __restrict__ s_ptr, float* __restrict__ out,
                           long long n) {
    long long t = (long long)blockIdx.x * blockDim.x + threadIdx.x;
    if (t < n) out[t] = a[t] + s_ptr[0] * b[t];
}

// ---------------------------------------------------------------- SpMM scatter
// agg[row[e]*D + d] += elem[e] * x[col[e]*D + d]
__global__ void spmm_scatter(const int* __restrict__ row, const int* __restrict__ col,
                             const float* __restrict__ elem, const float* __restrict__ x,
                             float* __restrict__ agg) {
    long long t = (long long)blockIdx.x * blockDim.x + threadIdx.x;
    if (t >= (long long)NEDGES * D) return;
    int e = (int)(t >> 7);
    int d = (int)(t & 127);
    int r = row[e];
    int c = col[e];
    atomicAdd(agg + (size_t)r * D + d, elem[e] * x[(size_t)c * D + d]);
}

// ---------------------------------------------------------------- fp32 WMMA GEMM
// Y[m, n] = act( sum_k X[m,k] * W[k,n] + bias[n] ),  X: nrows x 128, W: 128 x 128.
// Block: 256 threads = 8 waves; block covers 16 rows, wave w covers cols [16w, 16w+16).
// Uses V_WMMA_F32_16X16X4_F32: A 16x4 (2 VGPR), B 4x16 (2 VGPR), C/D 16x16 (8 VGPR).
template <bool RELU>
__global__ void gemm128_wmma(const float* __restrict__ X, const float* __restrict__ W,
                             const float* __restrict__ bias, float* __restrict__ Y) {
    __shared__ float Atile[16 * D];   // 8 KB
    const int tid  = threadIdx.x;
    const int wave = tid >> 5;
    const int lane = tid & 31;
    const int m0   = blockIdx.x * 16;

    // cooperative, coalesced A-tile load: 2048 floats / 256 threads
    const float* Xblk = X + (size_t)m0 * D;
#pragma unroll
    for (int i = 0; i < 8; ++i) {
        int idx = tid + i * 256;
        Atile[idx] = Xblk[idx];
    }
    __syncthreads();

    const int n     = wave * 16 + (lane & 15);  // output column
    const int mA    = lane & 15;                // A row held by this lane
    const int khalf = (lane >> 4) * 2;          // lanes 0-15: K {0,1}; 16-31: K {2,3}

    v8f c = {};
#pragma unroll 8
    for (int k0 = 0; k0 < D; k0 += 4) {
        v2f a, b;
        a.x = Atile[mA * D + k0 + khalf];
        a.y = Atile[mA * D + k0 + khalf + 1];
        b.x = W[(size_t)(k0 + khalf) * D + n];
        b.y = W[(size_t)(k0 + khalf + 1) * D + n];
        c = __builtin_amdgcn_wmma_f32_16x16x4_f32(false, a, false, b,
                                                  (short)0, c, false, false);
    }

    const float bv = bias[n];
#pragma unroll
    for (int v = 0; v < 8; ++v) {
        int m = m0 + v + 8 * (lane >> 4);
        float val = c[v] + bv;
        if (RELU) val = val > 0.0f ? val : 0.0f;
        Y[(size_t)m * D + n] = val;
    }
}

// ---------------------------------------------------------------- BatchNorm stats
// stats[0:128] += column sums, stats[128:256] += column sums of squares
__global__ void bn_stats(const float* __restrict__ z, float* __restrict__ stats,
                         int nrows, int rows_per_block) {
    int d  = threadIdx.x;  // 128 threads
    int r0 = blockIdx.x * rows_per_block;
    int r1 = r0 + rows_per_block; if (r1 > nrows) r1 = nrows;
    float s = 0.0f, sq = 0.0f;
    for (int r = r0; r < r1; ++r) {
        float v = z[(size_t)r * D + d];
        s += v; sq += v * v;
    }
    atomicAdd(stats + d, s);
    atomicAdd(stats + D + d, sq);
}

// h = leaky_relu( gamma*(z - mu)*rsqrt(var + eps) + beta )
__global__ void bn_norm_lrelu(const float* __restrict__ z, const float* __restrict__ stats,
                              const float* __restrict__ gamma, const float* __restrict__ beta,
                              float* __restrict__ h, long long n, float inv_n) {
    long long t = (long long)blockIdx.x * blockDim.x + threadIdx.x;
    if (t >= n) return;
    int d = (int)(t & 127);
    float mu  = stats[d] * inv_n;
    float var = stats[D + d] * inv_n - mu * mu;
    float v = gamma[d] * (z[t] - mu) * rsqrtf(var + 1e-5f) + beta[d];
    h[t] = LRELU(v);
}

// ---------------------------------------------------------------- pooling: attention
// one wave (32 lanes) per node: 128-wide dot via float4/lane + shuffle reduce
__global__ void pool_att(const float* __restrict__ hl, const float* __restrict__ gpW,
                         const float* __restrict__ gpb, const float* __restrict__ gp_in,
                         float* __restrict__ gp_out, int n) {
    int gt   = blockIdx.x * blockDim.x + threadIdx.x;
    int node = gt >> 5;
    int lane = threadIdx.x & 31;
    if (node >= n) return;
    const float* hp = hl + (size_t)node * D + lane * 4;
    const float* wp = gpW + lane * 4;
    float p = hp[0] * wp[0] + hp[1] * wp[1] + hp[2] * wp[2] + hp[3] * wp[3];
#pragma unroll
    for (int off = 16; off > 0; off >>= 1) p += __shfl_down(p, off, 32);
    if (lane == 0) {
        float att = p + gp_in[node] * gpW[D] + gpb[0];
        float lr  = LRELU(att);
        gp_out[node] = expf(-lr * 0.05f);
    }
}

// ---------------------------------------------------------------- pooling: segment sums
// node2graph is sorted: run-length accumulate, flush with one atomic per run
__global__ void pool_segsum(const float* __restrict__ hl, const float* __restrict__ gp,
                            const int* __restrict__ n2g, float* __restrict__ pooled,
                            float* __restrict__ rowsum, int n, int chunk) {
    int d  = threadIdx.x;  // 128 threads, thread = feature
    int i0 = blockIdx.x * chunk;
    int i1 = i0 + chunk; if (i1 > n) i1 = n;
    if (i0 >= i1) return;
    int curg = n2g[i0];
    float acc = 0.0f, accs = 0.0f;
    for (int i = i0; i < i1; ++i) {
        int g = n2g[i];
        if (g != curg) {
            atomicAdd(pooled + (size_t)curg * D + d, acc);
            if (d == 0) atomicAdd(rowsum + curg, accs);
            acc = 0.0f; accs = 0.0f; curg = g;
        }
        float w = gp[i];
        acc += w * hl[(size_t)i * D + d];
        if (d == 0) accs += w;
    }
    atomicAdd(pooled + (size_t)curg * D + d, acc);
    if (d == 0) atomicAdd(rowsum + curg, accs);
}

// ---------------------------------------------------------------- pooling: score GEMM
// out[g,o] += (pooled[g,:]/rowsum[g]) @ predW[:,o] + predb[o]
__global__ void pool_score(const float* __restrict__ pooled, const float* __restrict__ rowsum,
                           const float* __restrict__ predW, const float* __restrict__ predb,
                           float* __restrict__ out) {
    int g = blockIdx.x;    // 64
    int o = threadIdx.x;   // 64
    float inv = 1.0f / rowsum[g];
    float acc = 0.0f;
#pragma unroll 8
    for (int dd = 0; dd < D; ++dd)
        acc += pooled[(size_t)g * D + dd] * predW[(size_t)dd * NOUT + o];
    out[(size_t)g * NOUT + o] += acc * inv + predb[o];
}

// ---------------------------------------------------------------- driver
extern "C" void kernel_launch(void* const* d_in, const int* in_sizes, int n_in,
                              void* d_out, int out_size, void* d_ws, size_t ws_size,
                              hipStream_t stream) {
    const float* h0       = (const float*)d_in[0];
    const float* pos_enc  = (const float*)d_in[1];
    const float* edge_elem= (const float*)d_in[2];
    const float* elem_gp0 = (const float*)d_in[3];
    const float* pos      = (const float*)d_in[4];
    const float* gnn_W1   = (const float*)d_in[5];
    const float* gnn_b1   = (const float*)d_in[6];
    const float* gnn_W2   = (const float*)d_in[7];
    const float* gnn_b2   = (const float*)d_in[8];
    const float* bn_gamma = (const float*)d_in[9];
    const float* bn_beta  = (const float*)d_in[10];
    const float* gp_W     = (const float*)d_in[11];  // (3, 129)
    const float* gp_b     = (const float*)d_in[12];  // (3,)
    const float* pred_W   = (const float*)d_in[13];  // (3, 128, 64)
    const float* pred_b   = (const float*)d_in[14];  // (3, 64)
    const int*   edge_idx = (const int*)d_in[15];    // (2, E)
    const int*   n2g      = (const int*)d_in[16];    // (N,)
    float* out = (float*)d_out;

    const int* erow = edge_idx;
    const int* ecol = edge_idx + NEDGES;

    const long long NF = (long long)NNODES * D;  // 12.8M floats
    float* ws     = (float*)d_ws;
    float* hrep0  = ws;             // N x D
    float* h1     = ws + 1 * NF;    // N x D
    float* h2     = ws + 2 * NF;    // N x D
    float* xbuf   = ws + 3 * NF;    // N x D   (reused as z1)
    float* agg    = ws + 4 * NF;    // N x D   (reused as z)
    float* gpbuf  = ws + 5 * NF;    // N
    float* stats  = gpbuf + NNODES; // 256
    float* pooled = stats + 2 * D;  // 64 x 128
    float* rowsum = pooled + NGRAPH * D;  // 64

    const int TB = 256;
    const long long nfBlocks   = (NF + TB - 1) / TB;
    const long long edgeBlocks = ((long long)NEDGES * D + TB - 1) / TB;

    // zero output accumulator
    zero_f32<<<(NGRAPH * NOUT + TB - 1) / TB, TB, 0, stream>>>(out, NGRAPH * NOUT);

    // hidden_rep[0] = h0 + pos_enc   (scale = elem_gp0[0] == 1.0, but use direct add)
    // implemented as add_scaled with s from a ones array would be wrong; do explicit 1.0:
    // reuse add_scaled with elem_gp0 (all ones) as the scalar source.
    add_scaled<<<nfBlocks, TB, 0, stream>>>(h0, pos_enc, elem_gp0, hrep0, NF);

    float* hcur = const_cast<float*>(h0);
    float* hrep[3] = { hrep0, h1, h2 };

    for (int l = 0; l < 2; ++l) {
        // x = h + pos[l] * pos_enc
        add_scaled<<<nfBlocks, TB, 0, stream>>>(hcur, pos_enc, pos + l, xbuf, NF);
        // agg = 0 ; scatter
        zero_f32<<<nfBlocks, TB, 0, stream>>>(agg, NF);
        spmm_scatter<<<edgeBlocks, TB, 0, stream>>>(erow, ecol, edge_elem, xbuf, agg);
        // z1 = relu(agg @ W1 + b1)  -> xbuf      (WMMA fp32)
        gemm128_wmma<true><<<NNODES / 16, TB, 0, stream>>>(
            agg, gnn_W1 + (size_t)l * D * D, gnn_b1 + (size_t)l * D, xbuf);
        // z = z1 @ W2 + b2 -> agg                (WMMA fp32)
        gemm128_wmma<false><<<NNODES / 16, TB, 0, stream>>>(
            xbuf, gnn_W2 + (size_t)l * D * D, gnn_b2 + (size_t)l * D, agg);
        // BN stats + normalize + leaky_relu -> h_{l+1}
        zero_f32<<<1, TB, 0, stream>>>(stats, 2 * D);
        bn_stats<<<500, D, 0, stream>>>(agg, stats, NNODES, (NNODES + 499) / 500);
        bn_norm_lrelu<<<nfBlocks, TB, 0, stream>>>(agg, stats,
            bn_gamma + (size_t)l * D, bn_beta + (size_t)l * D, hrep[l + 1], NF,
            1.0f / (float)NNODES);
        hcur = hrep[l + 1];
    }

    // pooling heads (elem_gp chains sequentially across l)
    for (int l = 0; l < 3; ++l) {
        const float* hl    = hrep[l];
        const float* gp_in = (l == 0) ? elem_gp0 : gpbuf;
        zero_f32<<<(NGRAPH * D + NGRAPH + TB - 1) / TB, TB, 0, stream>>>(pooled,
                                                                         NGRAPH * D + NGRAPH);
        pool_att<<<(NNODES * 32 + TB - 1) / TB, TB, 0, stream>>>(
            hl, gp_W + (size_t)l * (D + 1), gp_b + l, gp_in, gpbuf, NNODES);
        pool_segsum<<<(NNODES + 255) / 256, D, 0, stream>>>(
            hl, gpbuf, n2g, pooled, rowsum, NNODES, 256);
        pool_score<<<NGRAPH, NOUT, 0, stream>>>(
            pooled, rowsum, pred_W + (size_t)l * D * NOUT, pred_b + (size_t)l * NOUT, out);
    }
}